// PAN_32298154066791
// MI455X (gfx1250) — compile-verified
//
#include <hip/hip_runtime.h>
#include <hip/hip_bf16.h>
#include <math.h>

// ---------------------------------------------------------------------------
// PAN forward for MI455X (gfx1250, wave32). All dense linear algebra runs on
// V_WMMA_F32_16X16X4_F32 (full f32 matrix pipe, matches reference numerics).
// LAE-conv is collapsed algebraically: softmax weights sum to 1, so
//   relu(sum_k a_k (g_k W + b)) == relu((sum_k a_k g_k) W + b)
// turning every neighborhood conv into one dense WMMA GEMM.
// ---------------------------------------------------------------------------

typedef float v2f __attribute__((ext_vector_type(2)));
typedef float v8f __attribute__((ext_vector_type(8)));

#define KNN 16

__device__ __forceinline__ int imin(int a, int b) { return a < b ? a : b; }

// ------------------------------ WMMA GEMM ----------------------------------
// C[b] = act( A[b] (MxK) * B[b] (KxN or (NxK)^T) + bias + Res[b] )
// One wave per 16x16 output tile. M must be a multiple of 16 (true for all
// call sites: B*N in {16384,2048,512,256}). N,K arbitrary.
// Main K-loop is completely guard-free (K rounded down to a multiple of 4);
// the single tail step uses clamped addresses + value selects (v_cndmask,
// no exec-mask divergence).
template <int TRANSB>
__global__ __launch_bounds__(32)
void gemm16_wmma_f32(const float* __restrict__ A, const float* __restrict__ Bm,
                     const float* __restrict__ bias, const float* __restrict__ Res,
                     float* __restrict__ C,
                     int M, int N, int K,
                     long long sA, long long sB, long long sC, long long sR,
                     int doRelu)
{
    const int lane = threadIdx.x & 31;
    const int li   = lane & 15;
    const int half = lane >> 4;
    const int tn   = blockIdx.x;
    const int tm   = blockIdx.y;
    const int bz   = blockIdx.z;

    const float* Ab = A  + (size_t)bz * sA;
    const float* Bb = Bm + (size_t)bz * sB;
    float*       Cb = C  + (size_t)bz * sC;
    const float* Rb = Res ? (Res + (size_t)bz * sR) : nullptr;

    const int row  = tm * 16 + li;          // A-frag row (M = li for both halves)
    const int col  = tn * 16 + li;          // B/C-frag column
    const int colc = imin(col, N - 1);      // clamped: loads always legal

    const float* Arow = Ab + (size_t)row * K;

    v8f acc = {0.f, 0.f, 0.f, 0.f, 0.f, 0.f, 0.f, 0.f};
    const int Kmain = K & ~3;

    if (TRANSB) {
        const float* Bcol = Bb + (size_t)colc * K;
#pragma unroll 4
        for (int k0 = 0; k0 < Kmain; k0 += 4) {
            const int ka = k0 + 2 * half;   // lane-half selects K pair {0,1}/{2,3}
            v2f a = *(const v2f*)(Arow + ka);
            v2f b = *(const v2f*)(Bcol + ka);
            acc = __builtin_amdgcn_wmma_f32_16x16x4_f32(false, a, false, b,
                                                        (short)0, acc, false, false);
        }
    } else {
#pragma unroll 4
        for (int k0 = 0; k0 < Kmain; k0 += 4) {
            const int ka = k0 + 2 * half;
            v2f a = *(const v2f*)(Arow + ka);
            v2f b;
            b.x = Bb[(size_t)ka * N + colc];
            b.y = Bb[(size_t)(ka + 1) * N + colc];
            acc = __builtin_amdgcn_wmma_f32_16x16x4_f32(false, a, false, b,
                                                        (short)0, acc, false, false);
        }
    }

    if (Kmain < K) {                        // tail (only K=3 at runtime)
        const int ka  = Kmain + 2 * half;
        const int k0c = imin(ka, K - 1);
        const int k1c = imin(ka + 1, K - 1);
        float ax = Arow[k0c], ay = Arow[k1c];
        float bx_, by_;
        if (TRANSB) {
            const float* Bcol = Bb + (size_t)colc * K;
            bx_ = Bcol[k0c]; by_ = Bcol[k1c];
        } else {
            bx_ = Bb[(size_t)k0c * N + colc];
            by_ = Bb[(size_t)k1c * N + colc];
        }
        v2f a, b;
        a.x = (ka     < K) ? ax  : 0.f;
        a.y = (ka + 1 < K) ? ay  : 0.f;
        b.x = (ka     < K) ? bx_ : 0.f;
        b.y = (ka + 1 < K) ? by_ : 0.f;
        acc = __builtin_amdgcn_wmma_f32_16x16x4_f32(false, a, false, b,
                                                    (short)0, acc, false, false);
    }

    if (col < N) {
        const float bv = bias ? bias[col] : 0.f;
        const int r0 = tm * 16 + half * 8;  // VGPR r -> M = half*8 + r
#pragma unroll
        for (int r = 0; r < 8; ++r) {
            const int rr = r0 + r;
            float v = acc[r] + bv;
            if (Rb) v += Rb[(size_t)rr * N + col];
            if (doRelu) v = fmaxf(v, 0.f);
            Cb[(size_t)rr * N + col] = v;
        }
    }
}

// --------------------------- ball query (top-16 NN) ------------------------
// Per-point insertion sort of (d2, idx); ties keep the lower index (matches
// jax.lax.top_k on -d2). Out-of-ball neighbors are replaced by self.
__global__ void ball_query_kernel(const float* __restrict__ xyz,
                                  int* __restrict__ nidx,
                                  int Bn, int N, float r2)
{
    int gid = blockIdx.x * blockDim.x + threadIdx.x;
    if (gid >= Bn * N) return;
    int b = gid / N, n = gid - b * N;
    const float* X = xyz + (size_t)b * N * 3;
    float px = X[3 * n], py = X[3 * n + 1], pz = X[3 * n + 2];

    float bd[KNN];
    int   bi[KNN];
#pragma unroll
    for (int k = 0; k < KNN; ++k) { bd[k] = 3.4e38f; bi[k] = 0; }

    for (int m = 0; m < N; ++m) {
        float dx = X[3 * m] - px, dy = X[3 * m + 1] - py, dz = X[3 * m + 2] - pz;
        float d = dx * dx + dy * dy + dz * dz;
        if (d < bd[KNN - 1]) {
            int j = KNN - 1;
            while (j > 0 && bd[j - 1] > d) { bd[j] = bd[j - 1]; bi[j] = bi[j - 1]; --j; }
            bd[j] = d; bi[j] = m;
        }
    }
    int* out = nidx + (size_t)gid * KNN;
#pragma unroll
    for (int k = 0; k < KNN; ++k) out[k] = (bd[k] <= r2) ? bi[k] : n;
}

// ------------------------ furthest point sampling --------------------------
// One workgroup per batch; LDS argmax with first-index tie-break (== argmax).
__global__ __launch_bounds__(256)
void fps_kernel(const float* __restrict__ xyz, int* __restrict__ out_idx,
                float* __restrict__ dist, int N, int S)
{
    const int b = blockIdx.x;
    const float* X = xyz + (size_t)b * N * 3;
    float* D = dist + (size_t)b * N;
    __shared__ float s_val[256];
    __shared__ int   s_idx[256];
    __shared__ int   s_far;
    const int t = threadIdx.x, nt = blockDim.x;

    for (int i = t; i < N; i += nt) D[i] = 1e10f;
    if (t == 0) s_far = 0;
    __syncthreads();

    for (int s = 0; s < S; ++s) {
        const int far = s_far;
        if (t == 0) out_idx[b * S + s] = far;
        const float cx = X[far * 3], cy = X[far * 3 + 1], cz = X[far * 3 + 2];
        float best = -1.f; int bi = 0x7fffffff;
        for (int i = t; i < N; i += nt) {
            float dx = X[i * 3] - cx, dy = X[i * 3 + 1] - cy, dz = X[i * 3 + 2] - cz;
            float nd = fminf(D[i], dx * dx + dy * dy + dz * dz);
            D[i] = nd;
            if (nd > best) { best = nd; bi = i; }
        }
        s_val[t] = best; s_idx[t] = bi;
        __syncthreads();
        for (int off = nt >> 1; off > 0; off >>= 1) {
            if (t < off) {
                if (s_val[t + off] > s_val[t] ||
                    (s_val[t + off] == s_val[t] && s_idx[t + off] < s_idx[t])) {
                    s_val[t] = s_val[t + off]; s_idx[t] = s_idx[t + off];
                }
            }
            __syncthreads();
        }
        if (t == 0) s_far = s_idx[0];
        __syncthreads();
    }
}

// ------------------------------- gather ------------------------------------
__global__ void gather_kernel(const float* __restrict__ src, const int* __restrict__ idx,
                              float* __restrict__ dst, int Bn, int Nsrc, int S, int C)
{
    long long gid = (long long)blockIdx.x * blockDim.x + threadIdx.x;
    long long tot = (long long)Bn * S * C;
    if (gid >= tot) return;
    int c = (int)(gid % C);
    long long bs = gid / C;
    int b = (int)(bs / S), s = (int)(bs - (long long)b * S);
    dst[gid] = src[((size_t)b * Nsrc + idx[b * S + s]) * C + c];
}

// --------- LAE prep: attention softmax over K + weighted feature sum -------
__global__ __launch_bounds__(128)
void lae_prep_kernel(const float* __restrict__ xyz, const float* __restrict__ fea,
                     const int* __restrict__ nidx, const float* __restrict__ Avec,
                     float* __restrict__ fbar, int N, int Cin)
{
    const int bn = blockIdx.x;
    const int b = bn / N, n = bn - b * N;
    __shared__ float alpha[KNN];
    __shared__ int   sidx[KNN];

    if (threadIdx.x == 0) {
        const float* X = xyz + (size_t)b * N * 3;
        const int* idx = nidx + (size_t)bn * KNN;
        const float A0 = Avec[0], A1 = Avec[1], A2 = Avec[2];
        const float px = X[3 * n], py = X[3 * n + 1], pz = X[3 * n + 2];
        float s[KNN], mx = -3.4e38f;
        for (int k = 0; k < KNN; ++k) {
            int m = idx[k];
            sidx[k] = m;
            float gx = X[3 * m] - px, gy = X[3 * m + 1] - py, gz = X[3 * m + 2] - pz;
            s[k] = gx * A0 + gy * A1 + gz * A2;
            mx = fmaxf(mx, s[k]);
        }
        float sum = 0.f;
        for (int k = 0; k < KNN; ++k) { float e = expf(s[k] - mx); alpha[k] = e; sum += e; }
        float inv = 1.f / sum;
        for (int k = 0; k < KNN; ++k) alpha[k] *= inv;
    }
    __syncthreads();

    const float* F = fea + (size_t)b * N * Cin;
    float* Out = fbar + (size_t)bn * Cin;
    for (int c = threadIdx.x; c < Cin; c += blockDim.x) {
        float acc = 0.f;
#pragma unroll
        for (int k = 0; k < KNN; ++k)
            acc += alpha[k] * F[(size_t)sidx[k] * Cin + c];
        Out[c] = acc;
    }
}

// --------------------------- row softmax (len <= 128) ----------------------
__global__ __launch_bounds__(64)
void softmax_rows_kernel(float* __restrict__ x, int rows, int L)
{
    int r = blockIdx.x;
    if (r >= rows) return;
    float* p = x + (size_t)r * L;
    __shared__ float red[64];
    const int t = threadIdx.x;

    float mx = -3.4e38f;
    for (int i = t; i < L; i += 64) mx = fmaxf(mx, p[i]);
    red[t] = mx; __syncthreads();
    for (int o = 32; o > 0; o >>= 1) { if (t < o) red[t] = fmaxf(red[t], red[t + o]); __syncthreads(); }
    mx = red[0]; __syncthreads();

    float sm = 0.f;
    for (int i = t; i < L; i += 64) { float e = expf(p[i] - mx); p[i] = e; sm += e; }
    red[t] = sm; __syncthreads();
    for (int o = 32; o > 0; o >>= 1) { if (t < o) red[t] += red[t + o]; __syncthreads(); }
    const float inv = 1.f / red[0];
    for (int i = t; i < L; i += 64) p[i] *= inv;
}

// ------------------------ 3-NN for feature propagation ---------------------
__global__ void knn3_kernel(const float* __restrict__ xd, const float* __restrict__ xs,
                            float* __restrict__ w3, int* __restrict__ i3,
                            int Bn, int Nd, int Ns)
{
    int gid = blockIdx.x * blockDim.x + threadIdx.x;
    if (gid >= Bn * Nd) return;
    int b = gid / Nd, n = gid - b * Nd;
    const float* Xd = xd + (size_t)b * Nd * 3;
    const float* Xs = xs + (size_t)b * Ns * 3;
    float px = Xd[3 * n], py = Xd[3 * n + 1], pz = Xd[3 * n + 2];
    float d0 = 3.4e38f, d1 = 3.4e38f, d2 = 3.4e38f;
    int   j0 = 0, j1 = 0, j2 = 0;
    for (int m = 0; m < Ns; ++m) {
        float dx = Xs[3 * m] - px, dy = Xs[3 * m + 1] - py, dz = Xs[3 * m + 2] - pz;
        float d = dx * dx + dy * dy + dz * dz;
        if (d < d0)      { d2 = d1; j2 = j1; d1 = d0; j1 = j0; d0 = d; j0 = m; }
        else if (d < d1) { d2 = d1; j2 = j1; d1 = d;  j1 = m; }
        else if (d < d2) { d2 = d;  j2 = m; }
    }
    float w0 = 1.f / fmaxf(d0, 1e-10f);
    float w1 = 1.f / fmaxf(d1, 1e-10f);
    float w2 = 1.f / fmaxf(d2, 1e-10f);
    float inv = 1.f / (w0 + w1 + w2);
    w3[gid * 3 + 0] = w0 * inv; w3[gid * 3 + 1] = w1 * inv; w3[gid * 3 + 2] = w2 * inv;
    i3[gid * 3 + 0] = j0;       i3[gid * 3 + 1] = j1;       i3[gid * 3 + 2] = j2;
}

// -------------------- interpolate + skip-concat ([fd | interp(fs)]) --------
__global__ void interp_concat_kernel(const float* __restrict__ fd, const float* __restrict__ fs,
                                     const float* __restrict__ w3, const int* __restrict__ i3,
                                     float* __restrict__ out,
                                     int Bn, int Nd, int Ns, int Cd, int Cs)
{
    const int C = Cd + Cs;
    long long gid = (long long)blockIdx.x * blockDim.x + threadIdx.x;
    long long tot = (long long)Bn * Nd * C;
    if (gid >= tot) return;
    int c = (int)(gid % C);
    long long bn = gid / C;
    int b = (int)(bn / Nd);
    if (c < Cd) {
        out[gid] = fd[(size_t)bn * Cd + c];
    } else {
        const int cc = c - Cd;
        const float* F = fs + (size_t)b * Ns * Cs;
        float acc = 0.f;
#pragma unroll
        for (int k = 0; k < 3; ++k)
            acc += w3[bn * 3 + k] * F[(size_t)i3[bn * 3 + k] * Cs + cc];
        out[gid] = acc;
    }
}

// --------------------------- (B,N,C) -> (B,C,N) ----------------------------
__global__ void transpose_out_kernel(const float* __restrict__ t, float* __restrict__ out,
                                     int Bn, int N, int C)
{
    long long gid = (long long)blockIdx.x * blockDim.x + threadIdx.x;
    long long tot = (long long)Bn * N * C;
    if (gid >= tot) return;
    int b = (int)(gid / ((long long)N * C));
    long long rem = gid - (long long)b * N * C;
    int n = (int)(rem / C), c = (int)(rem % C);
    out[((size_t)b * C + c) * N + n] = t[gid];
}

// ----------------------------- host-side helpers ---------------------------
static void launch_gemm(const float* A, const float* Bm, const float* bias, const float* Res,
                        float* C, int M, int N, int K,
                        long long sA, long long sB, long long sC, long long sR,
                        int transB, int doRelu, int batch, hipStream_t st)
{
    dim3 grid((N + 15) / 16, M / 16, batch);
    if (transB)
        gemm16_wmma_f32<1><<<grid, 32, 0, st>>>(A, Bm, bias, Res, C, M, N, K,
                                                sA, sB, sC, sR, doRelu);
    else
        gemm16_wmma_f32<0><<<grid, 32, 0, st>>>(A, Bm, bias, Res, C, M, N, K,
                                                sA, sB, sC, sR, doRelu);
}

static void run_lae(const float* xyzp, const float* feap, const int* nidx, const float* Avec,
                    const float* Wp, const float* bp, float* fbar, float* out,
                    int Bn, int Np, int Cin, int Cout, hipStream_t st)
{
    lae_prep_kernel<<<Bn * Np, 128, 0, st>>>(xyzp, feap, nidx, Avec, fbar, Np, Cin);
    launch_gemm(fbar, Wp, bp, nullptr, out, Bn * Np, Cout, Cin, 0, 0, 0, 0, 0, 1, 1, st);
}

static void run_attn(const float* fin, float* fout, const float* Wq, const float* Wk,
                     const float* Wv, float* q, float* k, float* v, float* s,
                     int Bn, int Np, int C, hipStream_t st)
{
    launch_gemm(fin, Wq, nullptr, nullptr, q, Bn * Np, C, C, 0, 0, 0, 0, 0, 0, 1, st);
    launch_gemm(fin, Wk, nullptr, nullptr, k, Bn * Np, C, C, 0, 0, 0, 0, 0, 0, 1, st);
    launch_gemm(fin, Wv, nullptr, nullptr, v, Bn * Np, C, C, 0, 0, 0, 0, 0, 0, 1, st);
    // scores = q @ k^T  (per batch)
    launch_gemm(q, k, nullptr, nullptr, s, Np, Np, C,
                (long long)Np * C, (long long)Np * C, (long long)Np * Np, 0, 1, 0, Bn, st);
    softmax_rows_kernel<<<Bn * Np, 64, 0, st>>>(s, Bn * Np, Np);
    // out = scores @ v + fin
    launch_gemm(s, v, nullptr, fin, fout, Np, C, Np,
                (long long)Np * Np, (long long)Np * C, (long long)Np * C,
                (long long)Np * C, 0, 0, Bn, st);
}

static void run_fp(const float* xd, const float* xs, const float* fd, const float* fs,
                   float* w3, int* i3, float* out, int Bn, int Nd, int Ns,
                   int Cd, int Cs, hipStream_t st)
{
    knn3_kernel<<<(Bn * Nd + 127) / 128, 128, 0, st>>>(xd, xs, w3, i3, Bn, Nd, Ns);
    long long tot = (long long)Bn * Nd * (Cd + Cs);
    interp_concat_kernel<<<(unsigned)((tot + 255) / 256), 256, 0, st>>>(fd, fs, w3, i3, out,
                                                                        Bn, Nd, Ns, Cd, Cs);
}

// ------------------------------ entry point --------------------------------
extern "C" void kernel_launch(void* const* d_in, const int* in_sizes, int n_in,
                              void* d_out, int out_size, void* d_ws, size_t ws_size,
                              hipStream_t stream)
{
    (void)in_sizes; (void)n_in; (void)out_size; (void)ws_size;
    const int Bn = 4, N0 = 4096, N1 = 512, N2 = 128, N3 = 64, NC = 13;

    const float* xyz = (const float*)d_in[0];
    const float *W[7], *bb[7], *Av[7];
    for (int i = 0; i < 7; ++i) {
        W[i]  = (const float*)d_in[1 + 3 * i];
        bb[i] = (const float*)d_in[2 + 3 * i];
        Av[i] = (const float*)d_in[3 + 3 * i];
    }
    const float *Wq[3], *Wk[3], *Wv[3];
    for (int j = 0; j < 3; ++j) {
        Wq[j] = (const float*)d_in[22 + 3 * j];
        Wk[j] = (const float*)d_in[23 + 3 * j];
        Wv[j] = (const float*)d_in[24 + 3 * j];
    }
    const float* Wf = (const float*)d_in[31];
    const float* bf = (const float*)d_in[32];

    // --- bump allocator over d_ws ---
    char* wsp = (char*)d_ws;
    size_t off = 0;
    auto AL = [&](size_t bytes) -> void* {
        void* p = wsp + off;
        off = (off + bytes + 255) & ~(size_t)255;
        return p;
    };
    int*   nidx0  = (int*)AL((size_t)Bn * N0 * KNN * 4);
    int*   nidx1  = (int*)AL((size_t)Bn * N1 * KNN * 4);
    int*   nidx2  = (int*)AL((size_t)Bn * N2 * KNN * 4);
    int*   nidx3  = (int*)AL((size_t)Bn * N3 * KNN * 4);
    int*   idx1   = (int*)AL((size_t)Bn * N1 * 4);
    int*   idx2   = (int*)AL((size_t)Bn * N2 * 4);
    int*   idx3   = (int*)AL((size_t)Bn * N3 * 4);
    float* fpsd   = (float*)AL((size_t)Bn * N0 * 4);
    float* xyz1   = (float*)AL((size_t)Bn * N1 * 3 * 4);
    float* xyz2   = (float*)AL((size_t)Bn * N2 * 3 * 4);
    float* xyz3   = (float*)AL((size_t)Bn * N3 * 3 * 4);
    float* fea0   = (float*)AL((size_t)Bn * N0 * 64 * 4);
    float* fbar   = (float*)AL((size_t)Bn * N0 * 320 * 4);  // max Cin at its level
    float* fea1g  = (float*)AL((size_t)Bn * N1 * 64 * 4);
    float* fea1   = (float*)AL((size_t)Bn * N1 * 128 * 4);
    float* fea2g  = (float*)AL((size_t)Bn * N2 * 128 * 4);
    float* fea2   = (float*)AL((size_t)Bn * N2 * 256 * 4);
    float* fea2a  = (float*)AL((size_t)Bn * N2 * 256 * 4);
    float* fea3g  = (float*)AL((size_t)Bn * N3 * 256 * 4);
    float* fea3   = (float*)AL((size_t)Bn * N3 * 512 * 4);
    float* fea3a  = (float*)AL((size_t)Bn * N3 * 512 * 4);
    float* qb     = (float*)AL((size_t)Bn * 128 * 512 * 4);
    float* kb     = (float*)AL((size_t)Bn * 128 * 512 * 4);
    float* vb     = (float*)AL((size_t)Bn * 128 * 512 * 4);
    float* sc     = (float*)AL((size_t)Bn * N2 * N2 * 4);
    float* fea2cat= (float*)AL((size_t)Bn * N2 * 768 * 4);
    float* fea2b  = (float*)AL((size_t)Bn * N2 * 256 * 4);
    float* fea2c  = (float*)AL((size_t)Bn * N2 * 256 * 4);
    float* fea1cat= (float*)AL((size_t)Bn * N1 * 384 * 4);
    float* fea1b  = (float*)AL((size_t)Bn * N1 * 256 * 4);
    float* fea0cat= (float*)AL((size_t)Bn * N0 * 320 * 4);
    float* fea0b  = (float*)AL((size_t)Bn * N0 * 128 * 4);
    float* outtmp = (float*)AL((size_t)Bn * N0 * NC * 4);
    float* w3     = (float*)AL((size_t)Bn * N0 * 3 * 4);
    int*   i3     = (int*)AL((size_t)Bn * N0 * 3 * 4);

    // ---- encoder ----
    ball_query_kernel<<<(Bn * N0 + 255) / 256, 256, 0, stream>>>(xyz, nidx0, Bn, N0, 0.06f * 0.06f);
    run_lae(xyz, xyz, nidx0, Av[0], W[0], bb[0], fbar, fea0, Bn, N0, 3, 64, stream);  // (B,N,64)

    fps_kernel<<<Bn, 256, 0, stream>>>(xyz, idx1, fpsd, N0, N1);
    gather_kernel<<<(Bn * N1 * 3 + 255) / 256, 256, 0, stream>>>(xyz,  idx1, xyz1,  Bn, N0, N1, 3);
    gather_kernel<<<(Bn * N1 * 64 + 255) / 256, 256, 0, stream>>>(fea0, idx1, fea1g, Bn, N0, N1, 64);
    ball_query_kernel<<<(Bn * N1 + 255) / 256, 256, 0, stream>>>(xyz1, nidx1, Bn, N1, 0.12f * 0.12f);
    run_lae(xyz1, fea1g, nidx1, Av[1], W[1], bb[1], fbar, fea1, Bn, N1, 64, 128, stream);  // (B,512,128)

    fps_kernel<<<Bn, 256, 0, stream>>>(xyz1, idx2, fpsd, N1, N2);
    gather_kernel<<<(Bn * N2 * 3 + 255) / 256, 256, 0, stream>>>(xyz1, idx2, xyz2,  Bn, N1, N2, 3);
    gather_kernel<<<(Bn * N2 * 128 + 255) / 256, 256, 0, stream>>>(fea1, idx2, fea2g, Bn, N1, N2, 128);
    ball_query_kernel<<<(Bn * N2 + 255) / 256, 256, 0, stream>>>(xyz2, nidx2, Bn, N2, 0.3f * 0.3f);
    run_lae(xyz2, fea2g, nidx2, Av[2], W[2], bb[2], fbar, fea2, Bn, N2, 128, 256, stream); // (B,128,256)
    run_attn(fea2, fea2a, Wq[0], Wk[0], Wv[0], qb, kb, vb, sc, Bn, N2, 256, stream);

    fps_kernel<<<Bn, 256, 0, stream>>>(xyz2, idx3, fpsd, N2, N3);
    gather_kernel<<<(Bn * N3 * 3 + 255) / 256, 256, 0, stream>>>(xyz2,  idx3, xyz3,  Bn, N2, N3, 3);
    gather_kernel<<<(Bn * N3 * 256 + 255) / 256, 256, 0, stream>>>(fea2a, idx3, fea3g, Bn, N2, N3, 256);
    ball_query_kernel<<<(Bn * N3 + 255) / 256, 256, 0, stream>>>(xyz3, nidx3, Bn, N3, 0.5f * 0.5f);
    run_lae(xyz3, fea3g, nidx3, Av[3], W[3], bb[3], fbar, fea3, Bn, N3, 256, 512, stream); // (B,64,512)
    run_attn(fea3, fea3a, Wq[1], Wk[1], Wv[1], qb, kb, vb, sc, Bn, N3, 512, stream);

    // ---- decoder ----
    run_fp(xyz2, xyz3, fea2a, fea3a, w3, i3, fea2cat, Bn, N2, N3, 256, 512, stream);      // (B,128,768)
    run_lae(xyz2, fea2cat, nidx2, Av[4], W[4], bb[4], fbar, fea2b, Bn, N2, 768, 256, stream);
    run_attn(fea2b, fea2c, Wq[2], Wk[2], Wv[2], qb, kb, vb, sc, Bn, N2, 256, stream);

    run_fp(xyz1, xyz2, fea1, fea2c, w3, i3, fea1cat, Bn, N1, N2, 128, 256, stream);       // (B,512,384)
    run_lae(xyz1, fea1cat, nidx1, Av[5], W[5], bb[5], fbar, fea1b, Bn, N1, 384, 256, stream);

    run_fp(xyz, xyz1, fea0, fea1b, w3, i3, fea0cat, Bn, N0, N1, 64, 256, stream);         // (B,4096,320)
    run_lae(xyz, fea0cat, nidx0, Av[6], W[6], bb[6], fbar, fea0b, Bn, N0, 320, 128, stream);

    // ---- classifier + transpose ----
    launch_gemm(fea0b, Wf, bf, nullptr, outtmp, Bn * N0, NC, 128, 0, 0, 0, 0, 0, 0, 1, stream);
    long long tot = (long long)Bn * N0 * NC;
    transpose_out_kernel<<<(unsigned)((tot + 255) / 256), 256, 0, stream>>>(outtmp, (float*)d_out,
                                                                            Bn, N0, NC);
}